// ChamferLoss_16003048145308
// MI455X (gfx1250) — compile-verified
//
#include <hip/hip_runtime.h>

typedef __attribute__((ext_vector_type(2))) float v2f;
typedef __attribute__((ext_vector_type(8))) float v8f;

#define N_PTS 8192
#define TILE  16
#define STRIPS (N_PTS / TILE)   // 512 target tiles per scan

// One wave (32 lanes) owns a 16-point "query" strip and scans all 8192
// "target" points in 16-point tiles.
//
// Homogeneous-coordinate trick: with K=4,
//   A[m][:] = (-2*qx, -2*qy, -2*qz, 1)
//   B[:][n] = ( tx,    ty,    tz,   ||t||^2 )
// => C[m][n] = ||t_n||^2 - 2*q_m.t_n   (one V_WMMA_F32_16X16X4_F32, no fixup)
// Running min over n kept in registers; sqrt deferred to the end:
//   min_dist[m] = sqrt(max(0, ||q_m||^2 + min_n C[m][n]))
__global__ __launch_bounds__(256) void ChamferLoss_wmma_kernel(
    const float* __restrict__ p1,
    const float* __restrict__ p2,
    float* __restrict__ out) {
  const int lane  = threadIdx.x & 31;
  const int wid   = (blockIdx.x * blockDim.x + threadIdx.x) >> 5;   // 0..4095
  const int dir   = wid >> 11;        // 0: query=p1 target=p2, 1: swapped
  const int rem   = wid & 2047;
  const int batch = rem >> 9;         // /512
  const int strip = rem & (STRIPS - 1);

  const float* __restrict__ q = dir ? p2 : p1;
  const float* __restrict__ t = dir ? p1 : p2;

  const int  lo    = lane & 15;
  const bool lower = lane < 16;

  // ---- A fragment: 16 query points, 32-bit 16x4 (MxK) layout ----
  // lanes 0-15: (K0,K1) = (-2qx,-2qy); lanes 16-31: (K2,K3) = (-2qz, 1)
  const float* qp = q + ((size_t)batch * N_PTS + (size_t)strip * TILE + lo) * 3;
  const float qx = qp[0], qy = qp[1], qz = qp[2];
  v2f a;
  a.x = lower ? (-2.0f * qx) : (-2.0f * qz);
  a.y = lower ? (-2.0f * qy) : 1.0f;

  // every lane has the full query point `lo` -> local squared norm, no shuffle
  const float sqq = fmaf(qx, qx, fmaf(qy, qy, qz * qz));

  // running min of (||t||^2 - 2*q.t) per C VGPR slot
  float acc[8];
#pragma unroll
  for (int j = 0; j < 8; ++j) acc[j] = 3.0e38f;

  const float* tbase = t + (size_t)batch * N_PTS * 3;

#pragma unroll 4
  for (int tile = 0; tile < STRIPS; ++tile) {
    const float* tp = tbase + ((size_t)tile * TILE + lo) * 3;
    const float tx = tp[0], ty = tp[1], tz = tp[2];
    const float sqt = fmaf(tx, tx, fmaf(ty, ty, tz * tz));
    // B fragment 4x16 (KxN): lanes 0-15 (K0,K1)=(tx,ty);
    //                        lanes 16-31 (K2,K3)=(tz,||t||^2)
    v2f b;
    b.x = lower ? tx : tz;
    b.y = lower ? ty : sqt;

    v8f c = {};
    c = __builtin_amdgcn_wmma_f32_16x16x4_f32(
        /*neg_a=*/false, a, /*neg_b=*/false, b,
        /*c_mod=*/(short)0, c, /*reuse_a=*/false, /*reuse_b=*/false);

#pragma unroll
    for (int j = 0; j < 8; ++j) acc[j] = fminf(acc[j], c[j]);
  }

  // ---- reduce min over N (16 lanes per half), finish per query point ----
  float sum = 0.0f;
#pragma unroll
  for (int j = 0; j < 8; ++j) {
    float m = acc[j];
    m = fminf(m, __shfl_xor(m, 1, 32));
    m = fminf(m, __shfl_xor(m, 2, 32));
    m = fminf(m, __shfl_xor(m, 4, 32));
    m = fminf(m, __shfl_xor(m, 8, 32));
    // lower half: row M=j ; upper half: row M=j+8. Lane with lo==M also
    // holds sqq of that query point.
    const float d2   = m + sqq;
    const float dist = sqrtf(fmaxf(d2, 0.0f));
    if ((lower && lo == j) || (!lower && lo == j + 8)) sum += dist;
  }

  // both directions are means over 4*8192 = 32768 points
  sum *= (1.0f / 32768.0f);

  // wave-wide sum, then one float atomic per wave
  sum += __shfl_xor(sum, 1, 32);
  sum += __shfl_xor(sum, 2, 32);
  sum += __shfl_xor(sum, 4, 32);
  sum += __shfl_xor(sum, 8, 32);
  sum += __shfl_xor(sum, 16, 32);
  if (lane == 0) atomicAdd(out, sum);
}

extern "C" void kernel_launch(void* const* d_in, const int* in_sizes, int n_in,
                              void* d_out, int out_size, void* d_ws, size_t ws_size,
                              hipStream_t stream) {
  (void)in_sizes; (void)n_in; (void)out_size; (void)d_ws; (void)ws_size;
  const float* p1 = (const float*)d_in[0];
  const float* p2 = (const float*)d_in[1];
  float* out = (float*)d_out;

  // harness poisons d_out; zero the accumulator (capture-legal async memset)
  hipMemsetAsync(out, 0, sizeof(float), stream);

  // 2 dirs * 4 batches * 512 strips = 4096 waves = 512 blocks * 8 waves
  ChamferLoss_wmma_kernel<<<dim3(512), dim3(256), 0, stream>>>(p1, p2, out);
}